// RootSGCNet_30683246363241
// MI455X (gfx1250) — compile-verified
//
#include <hip/hip_runtime.h>

typedef __attribute__((ext_vector_type(2))) float v2f;
typedef __attribute__((ext_vector_type(8))) float v8f;

#define NNODES 50000
#define NEDGES 800000
#define NF 96          // input features
#define NH 128         // hidden
#define NC 40          // classes
#define NCP 48         // classes padded to 3x16 WMMA tiles
#define NTILES (NNODES / 16)   // 3125 exact

// ---------------- utility kernels ----------------

__global__ void zero_kernel(float* __restrict__ p, int n4) {
    int i = blockIdx.x * blockDim.x + threadIdx.x;
    if (i < n4) ((float4*)p)[i] = make_float4(0.f, 0.f, 0.f, 0.f);
}

__global__ void degree_kernel(const int* __restrict__ dst, float* __restrict__ deg, int E) {
    int e = blockIdx.x * blockDim.x + threadIdx.x;
    if (e < E) unsafeAtomicAdd(&deg[dst[e]], 1.0f);
}

__global__ void norm_kernel(float* __restrict__ deg, int n) {
    int i = blockIdx.x * blockDim.x + threadIdx.x;
    if (i < n) {
        float d = deg[i];
        deg[i] = rsqrtf(d > 0.f ? d : 1.0f);
    }
}

// out[row][:] = in[row][:] * s[row]^pw   (float4 vectorized, NF=96 divisible by 4)
__global__ void scale_rows_kernel(float* __restrict__ outp, const float* __restrict__ inp,
                                  const float* __restrict__ s, int pw) {
    int i = blockIdx.x * blockDim.x + threadIdx.x;
    const int total = NNODES * (NF / 4);
    if (i >= total) return;
    int row = i / (NF / 4);
    float sc = s[row];
    if (pw == 2) sc *= sc;
    float4 v = ((const float4*)inp)[i];
    v.x *= sc; v.y *= sc; v.z *= sc; v.w *= sc;
    ((float4*)outp)[i] = v;
}

// edge-parallel scatter-add: hout[dst[e]] += hin[src[e]]; blockIdx.y selects 16-float chunk
__global__ void scatter_kernel(const int* __restrict__ src, const int* __restrict__ dst,
                               const float* __restrict__ hin, float* __restrict__ hout, int E) {
    int e = blockIdx.x * blockDim.x + threadIdx.x;
    if (e >= E) return;
    int c = blockIdx.y * 16;
    const float4* in4 = (const float4*)(hin + src[e] * NF + c);
    float* o = hout + dst[e] * NF + c;
#pragma unroll
    for (int i = 0; i < 4; ++i) {
        float4 v = in4[i];
        unsafeAtomicAdd(o + 4 * i + 0, v.x);
        unsafeAtomicAdd(o + 4 * i + 1, v.y);
        unsafeAtomicAdd(o + 4 * i + 2, v.z);
        unsafeAtomicAdd(o + 4 * i + 3, v.w);
    }
}

// Fold projection into the two linear layers and store in WMMA-fragment-
// interleaved layout: element (k, c) -> WT[which][ ((k>>1)*NCP + c)*2 + (k&1) ]
// so a lane's B fragment {B[k][c], B[k+1][c]} is one contiguous 64-bit LDS load.
//   which=0: sum_j Wproj[c][j]    * Wlin[j][k]
//   which=1: sum_j Wproj[c][NH+j] * Wsgc[j][k]
__global__ void fold_weights_kernel(const float* __restrict__ Wsgc, const float* __restrict__ Wlin,
                                    const float* __restrict__ Wproj, float* __restrict__ WT) {
    int idx = blockIdx.x * blockDim.x + threadIdx.x;
    if (idx >= 2 * NF * NCP) return;
    int which = idx / (NF * NCP);
    int r = idx - which * (NF * NCP);
    int k = r / NCP, c = r % NCP;
    float acc = 0.f;
    if (c < NC) {
        const float* W = which ? Wsgc : Wlin;          // [NH][NF]
        const float* P = Wproj + c * (2 * NH) + which * NH;
        for (int j = 0; j < NH; ++j) acc += P[j] * W[j * NF + k];
    }
    WT[which * (NF * NCP) + ((k >> 1) * NCP + c) * 2 + (k & 1)] = acc;
}

// ---------------- fused WMMA GEMM ----------------
// out[n][c] = sum_k feat[n][k]*Wc1[k][c] + (hraw[n][k]*nrm[n])*Wc2[k][c]
// One wave per 16-row tile; 3 N-tiles of 16 (NCP=48); K loop in steps of 4 via
// V_WMMA_F32_16X16X4_F32 (fp32-exact, matches reference precision).
__global__ __launch_bounds__(256) void gemm_kernel(const float* __restrict__ feat,
                                                   const float* __restrict__ hraw,
                                                   const float* __restrict__ nrm,
                                                   const float* __restrict__ WTg,
                                                   float* __restrict__ out) {
    __shared__ float WT[2 * NF * NCP];   // 36 KB, fragment-interleaved
    for (int i = threadIdx.x; i < 2 * NF * NCP; i += 256) WT[i] = WTg[i];
    __syncthreads();

    int wave = threadIdx.x >> 5;          // wave32
    int lane = threadIdx.x & 31;
    int tile = blockIdx.x * 8 + wave;
    if (tile >= NTILES) return;           // wave-uniform: EXEC all-ones for WMMA

    int half = lane >> 4;                 // 0: K pair {k0,k0+1}, 1: {k0+2,k0+3}
    int l15  = lane & 15;
    int rowA = tile * 16 + l15;
    float nr = nrm[rowA];
    const float* fRow = feat + rowA * NF;
    const float* hRow = hraw + rowA * NF;
    const float* W1 = WT;
    const float* W2 = WT + NF * NCP;

    v8f acc0 = {}; v8f acc1 = {}; v8f acc2 = {};

#pragma unroll 4
    for (int k0 = 0; k0 < NF; k0 += 4) {
        int ka = k0 + 2 * half;           // this lane's K offset (8B-aligned loads)
        v2f aF = *(const v2f*)(fRow + ka);
        v2f aH = *(const v2f*)(hRow + ka);
        aH = aH * nr;                     // fuse final norm scale into A fragment

        // fragment-interleaved: pair index = ka>>1, +16 cols => +32 elements
        int pb = ((ka >> 1) * NCP + l15) * 2;
        v2f b1n0 = *(const v2f*)(W1 + pb);
        v2f b1n1 = *(const v2f*)(W1 + pb + 32);
        v2f b1n2 = *(const v2f*)(W1 + pb + 64);
        v2f b2n0 = *(const v2f*)(W2 + pb);
        v2f b2n1 = *(const v2f*)(W2 + pb + 32);
        v2f b2n2 = *(const v2f*)(W2 + pb + 64);

        acc0 = __builtin_amdgcn_wmma_f32_16x16x4_f32(false, aF, false, b1n0, (short)0, acc0, false, false);
        acc1 = __builtin_amdgcn_wmma_f32_16x16x4_f32(false, aF, false, b1n1, (short)0, acc1, false, false);
        acc2 = __builtin_amdgcn_wmma_f32_16x16x4_f32(false, aF, false, b1n2, (short)0, acc2, false, false);
        acc0 = __builtin_amdgcn_wmma_f32_16x16x4_f32(false, aH, false, b2n0, (short)0, acc0, false, false);
        acc1 = __builtin_amdgcn_wmma_f32_16x16x4_f32(false, aH, false, b2n1, (short)0, acc1, false, false);
        acc2 = __builtin_amdgcn_wmma_f32_16x16x4_f32(false, aH, false, b2n2, (short)0, acc2, false, false);
    }

    // C/D layout: VGPR r, lanes 0-15 -> M=r, lanes 16-31 -> M=r+8; N=l15 within tile
#pragma unroll
    for (int r = 0; r < 8; ++r) {
        int row = tile * 16 + half * 8 + r;
        float* orow = out + row * NC;
        orow[l15]      = acc0[r];               // cols 0..15
        orow[16 + l15] = acc1[r];               // cols 16..31
        if (32 + l15 < NC) orow[32 + l15] = acc2[r];   // cols 32..39
    }
}

// ---------------- host launcher ----------------

extern "C" void kernel_launch(void* const* d_in, const int* in_sizes, int n_in,
                              void* d_out, int out_size, void* d_ws, size_t ws_size,
                              hipStream_t stream) {
    const float* feat  = (const float*)d_in[0];
    const int*   src   = (const int*)  d_in[1];
    const int*   dst   = (const int*)  d_in[2];
    const float* Wsgc  = (const float*)d_in[3];
    const float* Wlin  = (const float*)d_in[4];
    const float* Wproj = (const float*)d_in[5];
    float* out = (float*)d_out;

    float* ws  = (float*)d_ws;
    float* nrm = ws;                       // 50000 (padded to 50048)
    float* hA  = ws + 50048;               // 50000*96
    float* hB  = hA + NNODES * NF;         // 50000*96
    float* WT  = hB + NNODES * NF;         // 2*96*48 (fragment-interleaved)

    const int HVEC4 = NNODES * NF / 4;     // 1,200,000

    // folded weights (independent of graph work, launch first)
    fold_weights_kernel<<<(2 * NF * NCP + 255) / 256, 256, 0, stream>>>(Wsgc, Wlin, Wproj, WT);

    // degree -> norm
    zero_kernel<<<(50048 / 4 + 255) / 256, 256, 0, stream>>>(nrm, 50048 / 4);
    degree_kernel<<<(NEDGES + 255) / 256, 256, 0, stream>>>(dst, nrm, NEDGES);
    norm_kernel<<<(NNODES + 255) / 256, 256, 0, stream>>>(nrm, NNODES);

    // hop 1: hA = feat*norm ; hB = scatter(hA)
    scale_rows_kernel<<<(HVEC4 + 255) / 256, 256, 0, stream>>>(hA, feat, nrm, 1);
    zero_kernel<<<(HVEC4 + 255) / 256, 256, 0, stream>>>(hB, HVEC4);
    scatter_kernel<<<dim3(NEDGES / 256, NF / 16), 256, 0, stream>>>(src, dst, hA, hB, NEDGES);

    // hop 2: hA = hB*norm^2 ; hB = scatter(hA)   (trailing *norm fused into GEMM)
    scale_rows_kernel<<<(HVEC4 + 255) / 256, 256, 0, stream>>>(hA, hB, nrm, 2);
    zero_kernel<<<(HVEC4 + 255) / 256, 256, 0, stream>>>(hB, HVEC4);
    scatter_kernel<<<dim3(NEDGES / 256, NF / 16), 256, 0, stream>>>(src, dst, hA, hB, NEDGES);

    // fused dense head: out = feat@Wc1 + (hB*norm)@Wc2, WMMA fp32
    gemm_kernel<<<(NTILES + 7) / 8, 256, 0, stream>>>(feat, hB, nrm, WT, out);
}